// Model_80728205295637
// MI455X (gfx1250) — compile-verified
//
#include <hip/hip_runtime.h>

typedef __attribute__((ext_vector_type(2))) float v2f;
typedef __attribute__((ext_vector_type(8))) float v8f;

// ---------------------------------------------------------------------------
// Problem constants
// ---------------------------------------------------------------------------
#define NPTS 32768
#define NC0  2436
#define NC1  182
#define NC2  16
#define KNN0 64
#define KNN1 64
#define KNN2 32
#define NEGV (-3.0f)

// Padded decoder row counts (multiple of 16 so the WMMA GEMM needs no M guards)
#define NC1P 192    // pad of 182
#define NC0P 2448   // pad of 2436

// Workspace layout (offsets in floats)
static constexpr size_t OFF_SCAL  = 0;                              // 8 scalars (maxd2 per level)
static constexpr size_t OFF_MIND2 = 8;                              // 32768
static constexpr size_t OFF_FIDX0 = OFF_MIND2 + NPTS;               // 2436 i32
static constexpr size_t OFF_FIDX1 = OFF_FIDX0 + NC0;                // 182 i32
static constexpr size_t OFF_FIDX2 = OFF_FIDX1 + NC1;                // 16 i32
static constexpr size_t OFF_NN0   = OFF_FIDX2 + NC2;                // 2436*64 i32
static constexpr size_t OFF_VAL0  = OFF_NN0 + (size_t)NC0 * KNN0;   // 2436*64 i32
static constexpr size_t OFF_NN1   = OFF_VAL0 + (size_t)NC0 * KNN0;
static constexpr size_t OFF_VAL1  = OFF_NN1 + (size_t)NC1 * KNN1;
static constexpr size_t OFF_NN2   = OFF_VAL1 + (size_t)NC1 * KNN1;
static constexpr size_t OFF_VAL2  = OFF_NN2 + (size_t)NC2 * KNN2;
static constexpr size_t OFF_VAL3  = OFF_VAL2 + (size_t)NC2 * KNN2;  // 16 i32
static constexpr size_t OFF_CUR0  = OFF_VAL3 + 16;                  // 2436*131
static constexpr size_t OFF_CUR1  = OFF_CUR0 + (size_t)NC0 * 131;   // 182*259
static constexpr size_t OFF_CUR2  = OFF_CUR1 + (size_t)NC1 * 259;   // 16*515
static constexpr size_t OFF_GFEAT = OFF_CUR2 + (size_t)NC2 * 515;   // 1027
static constexpr size_t OFF_D3    = OFF_GFEAT + 1027;               // 16*512
static constexpr size_t OFF_D2    = OFF_D3 + 16 * 512;              // 182*256
static constexpr size_t OFF_D1    = OFF_D2 + (size_t)NC1 * 256;     // 2436*128
static constexpr size_t OFF_II    = OFF_D1 + (size_t)NC0 * 128;     // 32768*3 i32
static constexpr size_t OFF_IW    = OFF_II + (size_t)NPTS * 3;      // 32768*3
static constexpr size_t PINGF     = 2990080;                        // >= 11648*256 floats
static constexpr size_t OFF_A     = OFF_IW + (size_t)NPTS * 3;
static constexpr size_t OFF_B     = OFF_A + PINGF;

// ---------------------------------------------------------------------------
// Farthest point sampling: 1 block, sequential scan with block-wide argmax.
// Matches jnp scan: mind2=min(mind2,d2(last)); next=argmax(mind2) (first idx).
// ---------------------------------------------------------------------------
__global__ __launch_bounds__(1024)
void pn_fps(const float* __restrict__ pos, int stride, int N, int nSample,
            int* __restrict__ idx_out, float* __restrict__ mind2,
            float* __restrict__ maxd2_out)
{
    __shared__ float sv[1024];
    __shared__ int   si[1024];
    __shared__ float slast[3];
    const int tid = threadIdx.x;
    for (int i = tid; i < N; i += 1024) mind2[i] = 1e30f;
    if (tid == 0) {
        idx_out[0] = 0;
        slast[0] = pos[0]; slast[1] = pos[1]; slast[2] = pos[2];
    }
    __syncthreads();
    for (int s = 1; s < nSample; ++s) {
        const float lx = slast[0], ly = slast[1], lz = slast[2];
        float best = -1.0f; int bi = 0x7fffffff;
        for (int i = tid; i < N; i += 1024) {
            const float* p = pos + (size_t)i * stride;
            float dx = p[0] - lx, dy = p[1] - ly, dz = p[2] - lz;
            float d2 = dx * dx + dy * dy + dz * dz;
            float m = fminf(mind2[i], d2);
            mind2[i] = m;
            if (m > best) { best = m; bi = i; }   // strict > keeps first index
        }
        sv[tid] = best; si[tid] = bi;
        __syncthreads();
        for (int off = 512; off > 0; off >>= 1) {
            if (tid < off) {
                float ov = sv[tid + off]; int oi = si[tid + off];
                if (ov > sv[tid] || (ov == sv[tid] && oi < si[tid])) { sv[tid] = ov; si[tid] = oi; }
            }
            __syncthreads();
        }
        if (tid == 0) {
            int nxt = si[0];
            idx_out[s] = nxt;
            const float* p = pos + (size_t)nxt * stride;
            slast[0] = p[0]; slast[1] = p[1]; slast[2] = p[2];
        }
        __syncthreads();
    }
    // final min-update with last selected point, then max-reduce -> maxd2
    const float lx = slast[0], ly = slast[1], lz = slast[2];
    float best = -1.0f;
    for (int i = tid; i < N; i += 1024) {
        const float* p = pos + (size_t)i * stride;
        float dx = p[0] - lx, dy = p[1] - ly, dz = p[2] - lz;
        best = fmaxf(best, fminf(mind2[i], dx * dx + dy * dy + dz * dz));
    }
    sv[tid] = best;
    __syncthreads();
    for (int off = 512; off > 0; off >>= 1) {
        if (tid < off) sv[tid] = fmaxf(sv[tid], sv[tid + off]);
        __syncthreads();
    }
    if (tid == 0) maxd2_out[0] = sv[0];
}

// ---------------------------------------------------------------------------
// kNN per center: k iterations of lexicographic (d2, idx) argmin, excluding
// keys <= last selected key (no flag array needed). Writes nn, valid, cpos.
// ---------------------------------------------------------------------------
__global__ __launch_bounds__(256)
void pn_knn(const float* __restrict__ pts, int stride, int N,
            const int* __restrict__ cidx, int k,
            const float* __restrict__ maxd2_ptr,
            int* __restrict__ nn, int* __restrict__ valid,
            float* __restrict__ curnext, int curstride)
{
    __shared__ float sv[256];
    __shared__ int   si[256];
    __shared__ float sc[3];
    __shared__ float s_ld2;
    __shared__ int   s_li;
    const int m = blockIdx.x;
    const int tid = threadIdx.x;
    if (tid == 0) {
        int ci = cidx[m];
        sc[0] = pts[(size_t)ci * stride + 0];
        sc[1] = pts[(size_t)ci * stride + 1];
        sc[2] = pts[(size_t)ci * stride + 2];
        curnext[(size_t)m * curstride + 0] = sc[0];
        curnext[(size_t)m * curstride + 1] = sc[1];
        curnext[(size_t)m * curstride + 2] = sc[2];
        s_ld2 = -1.0f; s_li = -1;
    }
    __syncthreads();
    const float cx = sc[0], cy = sc[1], cz = sc[2];
    const float r2 = 4.0f * maxd2_ptr[0];   // radius = 2*maxd
    for (int j = 0; j < k; ++j) {
        const float ld2 = s_ld2; const int li = s_li;
        float best = 3.0e38f; int bi = 0x7fffffff;
        for (int i = tid; i < N; i += 256) {
            const float* p = pts + (size_t)i * stride;
            float dx = p[0] - cx, dy = p[1] - cy, dz = p[2] - cz;
            float d2 = dx * dx + dy * dy + dz * dz;
            bool gt = (d2 > ld2) || (d2 == ld2 && i > li);
            bool lt = (d2 < best) || (d2 == best && i < bi);
            if (gt && lt) { best = d2; bi = i; }
        }
        sv[tid] = best; si[tid] = bi;
        __syncthreads();
        for (int off = 128; off > 0; off >>= 1) {
            if (tid < off) {
                float ov = sv[tid + off]; int oi = si[tid + off];
                if (ov < sv[tid] || (ov == sv[tid] && oi < si[tid])) { sv[tid] = ov; si[tid] = oi; }
            }
            __syncthreads();
        }
        if (tid == 0) {
            nn[(size_t)m * k + j] = si[0];
            valid[(size_t)m * k + j] = (sv[0] <= r2) ? 1 : 0;
            s_ld2 = sv[0]; s_li = si[0];
        }
        __syncthreads();
    }
}

// ---------------------------------------------------------------------------
// Build grouped features gf[row, col]: rel position (3) + source features.
// ---------------------------------------------------------------------------
__global__ void pn_gather_gf(const int* __restrict__ nn, const int* __restrict__ cidx,
                             const float* __restrict__ pts, int stride,
                             int featcnt, int k, int centerStart, int centerCount,
                             float* __restrict__ dst)
{
    const int C = 3 + featcnt;
    size_t total = (size_t)centerCount * k * C;
    size_t t = (size_t)blockIdx.x * blockDim.x + threadIdx.x;
    if (t >= total) return;
    int    col = (int)(t % C);
    size_t row = t / C;
    int j  = (int)(row % k);
    int cl = (int)(row / k);
    int m  = centerStart + cl;
    int pidx = nn[(size_t)m * k + j];
    const float* src = pts + (size_t)pidx * stride;
    float v;
    if (col < 3) v = src[col] - pts[(size_t)cidx[m] * stride + col];
    else         v = src[3 + (col - 3)];
    dst[t] = v;
}

// ---------------------------------------------------------------------------
// GEMM + bias + tanh with f32 WMMA (V_WMMA_F32_16X16X4_F32).
// Wave computes 16(M) x 64(N).  PRECONDITION: M % 16 == 0, N % 64 == 0
// (call sites pad rows), so the main K loop and epilogue are guard-free;
// only the single K%4 tail step carries predication.  A/B row pointers are
// strength-reduced (advanced by 4 rows per step) to keep addressing off the
// load critical path.
// ---------------------------------------------------------------------------
__global__ __launch_bounds__(128)
void pn_gemm_tanh(const float* __restrict__ X, const float* __restrict__ W,
                  const float* __restrict__ bias, float* __restrict__ Y,
                  int M, int K, int N)
{
    const int lane = threadIdx.x & 31;
    const int wave = threadIdx.x >> 5;
    const int tileM = blockIdx.y * 4 + wave;
    if (tileM * 16 >= M) return;
    const int n0  = blockIdx.x * 64;
    const int row = lane & 15;
    const int kh  = lane >> 4;            // 0 or 1 (half-wave)
    const int r   = tileM * 16 + row;

    // A: this lane reads X[r, k0 + kh*2 .. +1]; B: W[k0 + kh*2 .. +1, n0+row+{0,16,32,48}]
    const float* xr = X + (size_t)r * K + kh * 2;
    const float* w0 = W + (size_t)(kh * 2) * N + n0 + row;
    const size_t wstep = (size_t)4 * N;

    v8f zero = {};
    v8f acc0 = zero, acc1 = zero, acc2 = zero, acc3 = zero;

    const int Kmain = K & ~3;
    int k0 = 0;
    for (; k0 < Kmain; k0 += 4) {
        v2f a;
        a.x = xr[0];
        a.y = xr[1];
        v2f b0, b1, b2, b3;
        b0.x = w0[0];  b0.y = w0[N + 0];
        b1.x = w0[16]; b1.y = w0[N + 16];
        b2.x = w0[32]; b2.y = w0[N + 32];
        b3.x = w0[48]; b3.y = w0[N + 48];
        xr += 4;
        w0 += wstep;
        acc0 = __builtin_amdgcn_wmma_f32_16x16x4_f32(false, a, false, b0, (short)0, acc0, false, false);
        acc1 = __builtin_amdgcn_wmma_f32_16x16x4_f32(false, a, false, b1, (short)0, acc1, false, false);
        acc2 = __builtin_amdgcn_wmma_f32_16x16x4_f32(false, a, false, b2, (short)0, acc2, false, false);
        acc3 = __builtin_amdgcn_wmma_f32_16x16x4_f32(false, a, false, b3, (short)0, acc3, false, false);
    }
    if (k0 < K) {   // K % 4 tail, executed at most once
        const int ka = k0 + kh * 2;
        const bool ok0 = (ka < K), ok1 = (ka + 1 < K);
        v2f a;
        a.x = ok0 ? xr[0] : 0.0f;
        a.y = ok1 ? xr[1] : 0.0f;
        v2f b0, b1, b2, b3;
        b0.x = ok0 ? w0[0]  : 0.0f;  b0.y = ok1 ? w0[N + 0]  : 0.0f;
        b1.x = ok0 ? w0[16] : 0.0f;  b1.y = ok1 ? w0[N + 16] : 0.0f;
        b2.x = ok0 ? w0[32] : 0.0f;  b2.y = ok1 ? w0[N + 32] : 0.0f;
        b3.x = ok0 ? w0[48] : 0.0f;  b3.y = ok1 ? w0[N + 48] : 0.0f;
        acc0 = __builtin_amdgcn_wmma_f32_16x16x4_f32(false, a, false, b0, (short)0, acc0, false, false);
        acc1 = __builtin_amdgcn_wmma_f32_16x16x4_f32(false, a, false, b1, (short)0, acc1, false, false);
        acc2 = __builtin_amdgcn_wmma_f32_16x16x4_f32(false, a, false, b2, (short)0, acc2, false, false);
        acc3 = __builtin_amdgcn_wmma_f32_16x16x4_f32(false, a, false, b3, (short)0, acc3, false, false);
    }

    // Epilogue: bias + tanh, unconditional stores (M multiple of 16).
    float* yb = Y + (size_t)(tileM * 16 + kh * 8) * N + n0 + row;
    const float bv0 = bias[n0 + row];
    const float bv1 = bias[n0 + 16 + row];
    const float bv2 = bias[n0 + 32 + row];
    const float bv3 = bias[n0 + 48 + row];
#pragma unroll
    for (int i = 0; i < 8; ++i) {
        float* yr = yb + (size_t)i * N;
        yr[0]  = tanhf(acc0[i] + bv0);
        yr[16] = tanhf(acc1[i] + bv1);
        yr[32] = tanhf(acc2[i] + bv2);
        yr[48] = tanhf(acc3[i] + bv3);
    }
}

// ---------------------------------------------------------------------------
// Masked max-pool over k neighbors -> cur[m, 3+c]
// ---------------------------------------------------------------------------
__global__ void pn_pool_max(const float* __restrict__ h, const int* __restrict__ valid,
                            int k, int centerStart, int count, int C,
                            float* __restrict__ cur, int curstride)
{
    size_t t = (size_t)blockIdx.x * blockDim.x + threadIdx.x;
    size_t total = (size_t)count * C;
    if (t >= total) return;
    int c  = (int)(t % C);
    int cl = (int)(t / C);
    int m  = centerStart + cl;
    float v = NEGV;
    for (int j = 0; j < k; ++j) {
        if (valid[(size_t)m * k + j])
            v = fmaxf(v, h[((size_t)cl * k + j) * C + c]);
    }
    cur[(size_t)m * curstride + 3 + c] = v;
}

// ---------------------------------------------------------------------------
// Encoder level 3: center = argmin ||p - mean||^2, build gf3 [16 x 515].
// ---------------------------------------------------------------------------
__global__ __launch_bounds__(256)
void pn_enc3_prep(const float* __restrict__ cur2, float* __restrict__ gfeat,
                  float* __restrict__ gf3, int* __restrict__ valid3)
{
    __shared__ float sc[3];
    if (threadIdx.x == 0) {
        float mx = 0, my = 0, mz = 0;
        for (int j = 0; j < 16; ++j) {
            mx += cur2[j * 515 + 0]; my += cur2[j * 515 + 1]; mz += cur2[j * 515 + 2];
        }
        mx *= (1.0f / 16.0f); my *= (1.0f / 16.0f); mz *= (1.0f / 16.0f);
        float bd = 3.0e38f; int bi = 0;
        for (int j = 0; j < 16; ++j) {
            float dx = cur2[j * 515 + 0] - mx;
            float dy = cur2[j * 515 + 1] - my;
            float dz = cur2[j * 515 + 2] - mz;
            float d2 = dx * dx + dy * dy + dz * dz;
            if (d2 < bd) { bd = d2; bi = j; }
        }
        sc[0] = cur2[bi * 515 + 0]; sc[1] = cur2[bi * 515 + 1]; sc[2] = cur2[bi * 515 + 2];
        gfeat[0] = sc[0]; gfeat[1] = sc[1]; gfeat[2] = sc[2];
    }
    __syncthreads();
    if (threadIdx.x < 16) valid3[threadIdx.x] = 1;
    for (int t = threadIdx.x; t < 16 * 515; t += 256) {
        int j = t / 515, col = t % 515;
        gf3[t] = (col < 3) ? (cur2[j * 515 + col] - sc[col]) : cur2[j * 515 + col];
    }
}

// Decoder level 3 input: [uk_pos | broadcast global feature] -> [16 x 1027]
__global__ void pn_dec3_buildX(const float* __restrict__ cur2,
                               const float* __restrict__ gfeat,
                               float* __restrict__ X)
{
    int t = blockIdx.x * blockDim.x + threadIdx.x;
    if (t >= 16 * 1027) return;
    int j = t / 1027, col = t % 1027;
    X[t] = (col < 3) ? cur2[j * 515 + col] : gfeat[col];
}

// ---------------------------------------------------------------------------
// 3-NN inverse-distance interpolation: select + weights (top_k semantics).
// ---------------------------------------------------------------------------
__global__ void pn_interp_select(const float* __restrict__ uk, int ustride, int M,
                                 const float* __restrict__ kp, int kstride, int Nk,
                                 int* __restrict__ ii, float* __restrict__ iw)
{
    int u = blockIdx.x * blockDim.x + threadIdx.x;
    if (u >= M) return;
    float ux = uk[(size_t)u * ustride + 0];
    float uy = uk[(size_t)u * ustride + 1];
    float uz = uk[(size_t)u * ustride + 2];
    float d0 = 3e38f, d1 = 3e38f, d2 = 3e38f;
    int   i0 = 0,     i1 = 0,     i2 = 0;
    for (int i = 0; i < Nk; ++i) {
        const float* p = kp + (size_t)i * kstride;
        float dx = p[0] - ux, dy = p[1] - uy, dz = p[2] - uz;
        float d = dx * dx + dy * dy + dz * dz;
        if (d < d0)      { d2 = d1; i2 = i1; d1 = d0; i1 = i0; d0 = d; i0 = i; }
        else if (d < d1) { d2 = d1; i2 = i1; d1 = d; i1 = i; }
        else if (d < d2) { d2 = d; i2 = i; }
    }
    float w0 = 1.0f / fmaxf(sqrtf(d0), 1e-10f);
    float w1 = 1.0f / fmaxf(sqrtf(d1), 1e-10f);
    float w2 = 1.0f / fmaxf(sqrtf(d2), 1e-10f);
    float wsum = w0 + w1 + w2;
    iw[(size_t)u * 3 + 0] = w0 / wsum;
    iw[(size_t)u * 3 + 1] = w1 / wsum;
    iw[(size_t)u * 3 + 2] = w2 / wsum;
    ii[(size_t)u * 3 + 0] = i0;
    ii[(size_t)u * 3 + 1] = i1;
    ii[(size_t)u * 3 + 2] = i2;
}

// Build decoder input rows: [uk_pos | sum_t w_t * kfeat[idx_t]].
// Writes `urows` rows (padded to a multiple of 16); rows >= ureal are zero.
__global__ void pn_interp_buildX(const float* __restrict__ uk, int ustride,
                                 const int* __restrict__ ii, const float* __restrict__ iw,
                                 const float* __restrict__ kfeat, int F,
                                 int ustart, int urows, int ureal,
                                 float* __restrict__ X)
{
    const int C = 3 + F;
    size_t t = (size_t)blockIdx.x * blockDim.x + threadIdx.x;
    if (t >= (size_t)urows * C) return;
    int col = (int)(t % C);
    int rl  = (int)(t / C);
    int u   = ustart + rl;
    float v = 0.0f;
    if (u < ureal) {
        if (col < 3) {
            v = uk[(size_t)u * ustride + col];
        } else {
            int c = col - 3;
            v = iw[(size_t)u * 3 + 0] * kfeat[(size_t)ii[(size_t)u * 3 + 0] * F + c]
              + iw[(size_t)u * 3 + 1] * kfeat[(size_t)ii[(size_t)u * 3 + 1] * F + c]
              + iw[(size_t)u * 3 + 2] * kfeat[(size_t)ii[(size_t)u * 3 + 2] * F + c];
        }
    }
    X[t] = v;
}

// ---------------------------------------------------------------------------
// Host-side launch helpers
// ---------------------------------------------------------------------------
static inline void launch_gemm(const float* X, const float* W, const float* b,
                               float* Y, int M, int K, int N, hipStream_t s)
{
    dim3 grid(N / 64, (((M + 15) / 16) + 3) / 4);
    pn_gemm_tanh<<<grid, 128, 0, s>>>(X, W, b, Y, M, K, N);
}

static inline int ewgrid(size_t n) { return (int)((n + 255) / 256); }

extern "C" void kernel_launch(void* const* d_in, const int* in_sizes, int n_in,
                              void* d_out, int out_size, void* d_ws, size_t ws_size,
                              hipStream_t stream)
{
    (void)in_sizes; (void)n_in; (void)out_size; (void)ws_size;

    const float* x = (const float*)d_in[0];
    const float *eW[4][3], *eB[4][3], *dW[4][2], *dB[4][2];
    int p = 1;
    for (int v = 0; v < 4; ++v)
        for (int l = 0; l < 3; ++l) { eW[v][l] = (const float*)d_in[p++]; eB[v][l] = (const float*)d_in[p++]; }
    for (int v = 0; v < 4; ++v)
        for (int l = 0; l < 2; ++l) { dW[v][l] = (const float*)d_in[p++]; dB[v][l] = (const float*)d_in[p++]; }

    float* wsf   = (float*)d_ws;
    float* scal  = wsf + OFF_SCAL;
    float* mind2 = wsf + OFF_MIND2;
    int*   fidx0 = (int*)(wsf + OFF_FIDX0);
    int*   fidx1 = (int*)(wsf + OFF_FIDX1);
    int*   fidx2 = (int*)(wsf + OFF_FIDX2);
    int*   nn0   = (int*)(wsf + OFF_NN0);
    int*   val0  = (int*)(wsf + OFF_VAL0);
    int*   nn1   = (int*)(wsf + OFF_NN1);
    int*   val1  = (int*)(wsf + OFF_VAL1);
    int*   nn2   = (int*)(wsf + OFF_NN2);
    int*   val2  = (int*)(wsf + OFF_VAL2);
    int*   val3  = (int*)(wsf + OFF_VAL3);
    float* cur0  = wsf + OFF_CUR0;   // [2436 x 131]
    float* cur1  = wsf + OFF_CUR1;   // [182 x 259]
    float* cur2  = wsf + OFF_CUR2;   // [16 x 515]
    float* gfeat = wsf + OFF_GFEAT;  // [1027]
    float* dec3o = wsf + OFF_D3;     // [16 x 512]
    float* dec2o = wsf + OFF_D2;     // [182 x 256]
    float* dec1o = wsf + OFF_D1;     // [2436 x 128]
    int*   ii    = (int*)(wsf + OFF_II);
    float* iw    = wsf + OFF_IW;
    float* A     = wsf + OFF_A;
    float* B     = wsf + OFF_B;
    float* out   = (float*)d_out;    // [32768 x 128]

    // ------------------- Encoder level 0 -------------------
    pn_fps<<<1, 1024, 0, stream>>>(x, 6, NPTS, NC0, fidx0, mind2, &scal[0]);
    pn_knn<<<NC0, 256, 0, stream>>>(x, 6, NPTS, fidx0, KNN0, &scal[0], nn0, val0, cur0, 131);
    for (int cs = 0; cs < NC0; cs += 256) {
        int cnt  = (NC0 - cs < 256) ? (NC0 - cs) : 256;
        int rows = cnt * KNN0;
        size_t tg = (size_t)rows * 6;
        pn_gather_gf<<<ewgrid(tg), 256, 0, stream>>>(nn0, fidx0, x, 6, 3, KNN0, cs, cnt, A);
        launch_gemm(A, eW[0][0], eB[0][0], B, rows, 6, 64, stream);
        launch_gemm(B, eW[0][1], eB[0][1], A, rows, 64, 64, stream);
        launch_gemm(A, eW[0][2], eB[0][2], B, rows, 64, 128, stream);
        size_t tp = (size_t)cnt * 128;
        pn_pool_max<<<ewgrid(tp), 256, 0, stream>>>(B, val0, KNN0, cs, cnt, 128, cur0, 131);
    }

    // ------------------- Encoder level 1 -------------------
    pn_fps<<<1, 1024, 0, stream>>>(cur0, 131, NC0, NC1, fidx1, mind2, &scal[1]);
    pn_knn<<<NC1, 256, 0, stream>>>(cur0, 131, NC0, fidx1, KNN1, &scal[1], nn1, val1, cur1, 259);
    {
        int rows = NC1 * KNN1;  // 11648
        size_t tg = (size_t)rows * 131;
        pn_gather_gf<<<ewgrid(tg), 256, 0, stream>>>(nn1, fidx1, cur0, 131, 128, KNN1, 0, NC1, A);
        launch_gemm(A, eW[1][0], eB[1][0], B, rows, 131, 128, stream);
        launch_gemm(B, eW[1][1], eB[1][1], A, rows, 128, 128, stream);
        launch_gemm(A, eW[1][2], eB[1][2], B, rows, 128, 256, stream);
        pn_pool_max<<<ewgrid((size_t)NC1 * 256), 256, 0, stream>>>(B, val1, KNN1, 0, NC1, 256, cur1, 259);
    }

    // ------------------- Encoder level 2 -------------------
    pn_fps<<<1, 1024, 0, stream>>>(cur1, 259, NC1, NC2, fidx2, mind2, &scal[2]);
    pn_knn<<<NC2, 256, 0, stream>>>(cur1, 259, NC1, fidx2, KNN2, &scal[2], nn2, val2, cur2, 515);
    {
        int rows = NC2 * KNN2;  // 512
        size_t tg = (size_t)rows * 259;
        pn_gather_gf<<<ewgrid(tg), 256, 0, stream>>>(nn2, fidx2, cur1, 259, 256, KNN2, 0, NC2, A);
        launch_gemm(A, eW[2][0], eB[2][0], B, rows, 259, 256, stream);
        launch_gemm(B, eW[2][1], eB[2][1], A, rows, 256, 256, stream);
        launch_gemm(A, eW[2][2], eB[2][2], B, rows, 256, 512, stream);
        pn_pool_max<<<ewgrid((size_t)NC2 * 512), 256, 0, stream>>>(B, val2, KNN2, 0, NC2, 512, cur2, 515);
    }

    // ------------------- Encoder level 3 (global) -------------------
    pn_enc3_prep<<<1, 256, 0, stream>>>(cur2, gfeat, A, val3);
    launch_gemm(A, eW[3][0], eB[3][0], B, 16, 515, 512, stream);
    launch_gemm(B, eW[3][1], eB[3][1], A, 16, 512, 512, stream);
    launch_gemm(A, eW[3][2], eB[3][2], B, 16, 512, 1024, stream);
    pn_pool_max<<<ewgrid(1024), 256, 0, stream>>>(B, val3, 16, 0, 1, 1024, gfeat, 1027);

    // ------------------- Decoder v=3 -------------------
    pn_dec3_buildX<<<ewgrid((size_t)16 * 1027), 256, 0, stream>>>(cur2, gfeat, A);
    launch_gemm(A, dW[0][0], dB[0][0], B, 16, 1027, 512, stream);
    launch_gemm(B, dW[0][1], dB[0][1], A, 16, 512, 512, stream);
    hipMemcpyAsync(dec3o, A, (size_t)16 * 512 * sizeof(float), hipMemcpyDeviceToDevice, stream);

    // ------------------- Decoder v=2 (rows padded 182 -> 192) -------------------
    pn_interp_select<<<ewgrid(NC1), 256, 0, stream>>>(cur1, 259, NC1, cur2, 515, 16, ii, iw);
    pn_interp_buildX<<<ewgrid((size_t)NC1P * 515), 256, 0, stream>>>(cur1, 259, ii, iw, dec3o, 512, 0, NC1P, NC1, A);
    launch_gemm(A, dW[1][0], dB[1][0], B, NC1P, 515, 512, stream);
    launch_gemm(B, dW[1][1], dB[1][1], A, NC1P, 512, 256, stream);
    hipMemcpyAsync(dec2o, A, (size_t)NC1 * 256 * sizeof(float), hipMemcpyDeviceToDevice, stream);

    // ------------------- Decoder v=1 (rows padded 2436 -> 2448) -------------------
    pn_interp_select<<<ewgrid(NC0), 256, 0, stream>>>(cur0, 131, NC0, cur1, 259, NC1, ii, iw);
    pn_interp_buildX<<<ewgrid((size_t)NC0P * 259), 256, 0, stream>>>(cur0, 131, ii, iw, dec2o, 256, 0, NC0P, NC0, A);
    launch_gemm(A, dW[2][0], dB[2][0], B, NC0P, 259, 256, stream);
    launch_gemm(B, dW[2][1], dB[2][1], A, NC0P, 256, 128, stream);
    hipMemcpyAsync(dec1o, A, (size_t)NC0 * 128 * sizeof(float), hipMemcpyDeviceToDevice, stream);

    // ------------------- Decoder v=0 (chunked, writes d_out) -------------------
    pn_interp_select<<<ewgrid(NPTS), 256, 0, stream>>>(x, 6, NPTS, cur0, 131, NC0, ii, iw);
    for (int us = 0; us < NPTS; us += 16384) {
        int cnt = 16384;
        pn_interp_buildX<<<ewgrid((size_t)cnt * 131), 256, 0, stream>>>(x, 6, ii, iw, dec1o, 128, us, cnt, NPTS, A);
        launch_gemm(A, dW[3][0], dB[3][0], B, cnt, 131, 128, stream);
        launch_gemm(B, dW[3][1], dB[3][1], out + (size_t)us * 128, cnt, 128, 128, stream);
    }
}